// HungarianMatcher_64244120813970
// MI455X (gfx1250) — compile-verified
//
#include <hip/hip_runtime.h>
#include <hip/hip_bf16.h>

typedef __attribute__((ext_vector_type(2))) float v2f;
typedef __attribute__((ext_vector_type(8))) float v8f;
typedef unsigned int u32x4 __attribute__((ext_vector_type(4)));
typedef int          i32x8 __attribute__((ext_vector_type(8)));
typedef int          i32x4 __attribute__((ext_vector_type(4)));

#define BQ 1024
#define QQ 128
#define TT 32
#define NC 257
#define NCP 260          // K padded to multiple of 4 for 16x16x4 WMMA
#define W_BBOX 5.0f
#define W_GIOU 2.0f
#define CINF 1e9f

// ---------------------------------------------------------------------------
// Kernel 1: cost matrix. One block (256 threads, 8 waves) per batch element.
// - logits tile staged into LDS by the Tensor Data Mover (tensor_load_to_lds,
//   TENSORcnt). TDM OOB-zero semantics provide the K padding (257 -> 260).
// - cost_class computed as probs @ onehot(labels) with V_WMMA_F32_16X16X4_F32:
//   the one-hot makes the WMMA result bit-exact equal to the gather.
// ---------------------------------------------------------------------------
__global__ __launch_bounds__(256) void cost_kernel(
    const float* __restrict__ logits,   // [B,Q,257]
    const float* __restrict__ pboxes,   // [B,Q,4]
    const int*   __restrict__ labels,   // [B,T]
    const float* __restrict__ tboxes,   // [B,T,4]
    float*       __restrict__ Cout)     // [B,Q,T]
{
    __shared__ float E[NCP][TT];        // one-hot  [K][N]  (33 KB)
    __shared__ float P[16][NCP];        // logits->probs tile [M][K] (16.6 KB)
    __shared__ float Ccls[16][TT];      // gathered probs (WMMA result)
    __shared__ float Tb[TT][4];
    __shared__ int   Lab[TT];

    const int b    = blockIdx.x;
    const int tid  = threadIdx.x;
    const int lane = tid & 31;
    const int wave = tid >> 5;

    // --- targets + one-hot build ---
    if (tid < TT) {
        Lab[tid] = labels[b * TT + tid];
        #pragma unroll
        for (int k = 0; k < 4; ++k) Tb[tid][k] = tboxes[(b * TT + tid) * 4 + k];
    }
    for (int idx = tid; idx < NCP * TT; idx += 256) ((float*)E)[idx] = 0.0f;
    __syncthreads();
    if (tid < TT) E[Lab[tid]][tid] = 1.0f;

    const unsigned lds_P = (unsigned)(unsigned long long)(void*)&P[0][0];
    const int r  = tid >> 4;   // row within 16-row tile
    const int c0 = tid & 15;   // 16 threads cooperate per row

    for (int qt = 0; qt < QQ / 16; ++qt) {
        // ---- TDM: DMA the 16x257 logits tile into P (stride 260, OOB->0) ----
        if (wave == 0) {
            const unsigned long long gaddr =
                (unsigned long long)(logits + ((size_t)b * QQ + qt * 16) * NC);
            u32x4 g0;
            g0[0] = 1u;                                   // count=1 (valid D#)
            g0[1] = lds_P;                                // lds_addr
            g0[2] = (unsigned)(gaddr & 0xffffffffu);      // global_addr[31:0]
            g0[3] = (unsigned)((gaddr >> 32) & 0x1ffffffu) | (2u << 30); // [56:32] | type=2
            i32x8 g1;
            g1[0] = 0x2 << 16;                            // data_size=2 (4 bytes)
            g1[1] = (int)((NC & 0xffff) << 16);           // tensor_dim0[15:0]=257
            g1[2] = 16 << 16;                             // tensor_dim1=16
            g1[3] = NCP << 16;                            // tile_dim0=260 (OOB cols read 0)
            g1[4] = 16;                                   // tile_dim1=16
            g1[5] = NC;                                   // tensor_dim0_stride=257
            g1[6] = 0;
            g1[7] = 0;
            i32x4 gz4 = {0, 0, 0, 0};                     // groups 2/3 unused (2-D tile)
            i32x8 gz8 = {0, 0, 0, 0, 0, 0, 0, 0};         // trailing group (clang-23 6-arg form)
            __builtin_amdgcn_tensor_load_to_lds(g0, g1, gz4, gz4, gz8, 0);
            __builtin_amdgcn_s_wait_tensorcnt(0);
        }
        if (qt + 1 < QQ / 16)   // gfx1250 global_prefetch_b8 of next tile
            __builtin_prefetch(logits + ((size_t)b * QQ + (qt + 1) * 16 + r) * NC + c0, 0, 0);
        __syncthreads();        // publish TDM tile to all waves

        // softmax (max-subtracted, f32) in place on the P tile
        float m = -CINF;
        for (int c = c0; c < NC; c += 16) m = fmaxf(m, P[r][c]);
        #pragma unroll
        for (int o = 8; o >= 1; o >>= 1) m = fmaxf(m, __shfl_xor(m, o, 32));
        float s = 0.0f;
        for (int c = c0; c < NC; c += 16) { float e = expf(P[r][c] - m); P[r][c] = e; s += e; }
        #pragma unroll
        for (int o = 8; o >= 1; o >>= 1) s += __shfl_xor(s, o, 32);
        const float inv = 1.0f / s;
        for (int c = c0; c < NC; c += 16) P[r][c] *= inv;   // pad cols stay 0 from TDM
        __syncthreads();

        // WMMA gather: waves 0/1 each produce one 16x16 N-tile of probs@onehot
        if (wave < 2) {
            const int n0    = wave * 16;
            const int mrow  = lane & 15;             // A: M index / B: N index
            const int khalf = (lane >> 4) << 1;      // K sub-offset 0 or 2
            v8f acc = {};
            for (int k0 = 0; k0 < NCP; k0 += 4) {
                v2f a, bm;
                a[0]  = P[mrow][k0 + khalf];
                a[1]  = P[mrow][k0 + khalf + 1];
                bm[0] = E[k0 + khalf][n0 + mrow];
                bm[1] = E[k0 + khalf + 1][n0 + mrow];
                acc = __builtin_amdgcn_wmma_f32_16x16x4_f32(
                          false, a, false, bm, (short)0, acc, false, false);
            }
            #pragma unroll
            for (int rr = 0; rr < 8; ++rr)           // D: M = rr + 8*(lane>=16)
                Ccls[rr + ((lane >> 4) << 3)][n0 + (lane & 15)] = acc[rr];
        }
        __syncthreads();

        // combine: 5*L1 - prob - 2*giou   (512 elems, 2 per thread)
        for (int e = tid; e < 16 * TT; e += 256) {
            const int rr = e >> 5, t = e & 31;
            const int q2 = qt * 16 + rr;
            const float* pb = pboxes + ((size_t)b * QQ + q2) * 4;
            const float cx = pb[0], cy = pb[1], w = pb[2], h = pb[3];
            const float tx = Tb[t][0], ty = Tb[t][1], tw = Tb[t][2], th = Tb[t][3];
            const float l1 = fabsf(cx - tx) + fabsf(cy - ty) + fabsf(w - tw) + fabsf(h - th);
            const float p1x = cx - 0.5f * w,  p1y = cy - 0.5f * h;
            const float p2x = cx + 0.5f * w,  p2y = cy + 0.5f * h;
            const float q1x = tx - 0.5f * tw, q1y = ty - 0.5f * th;
            const float q2x = tx + 0.5f * tw, q2y = ty + 0.5f * th;
            const float area1 = (p2x - p1x) * (p2y - p1y);
            const float area2 = (q2x - q1x) * (q2y - q1y);
            const float iw = fmaxf(fminf(p2x, q2x) - fmaxf(p1x, q1x), 0.0f);
            const float ih = fmaxf(fminf(p2y, q2y) - fmaxf(p1y, q1y), 0.0f);
            const float inter = iw * ih;
            const float uni   = area1 + area2 - inter;
            const float iou   = inter / uni;
            const float cw = fmaxf(fmaxf(p2x, q2x) - fminf(p1x, q1x), 0.0f);
            const float ch = fmaxf(fmaxf(p2y, q2y) - fminf(p1y, q1y), 0.0f);
            const float areac = cw * ch;
            const float giou  = iou - (areac - uni) / (areac + 1e-6f);
            Cout[((size_t)b * QQ + q2) * TT + t] = W_BBOX * l1 - Ccls[rr][t] - W_GIOU * giou;
        }
        __syncthreads();
    }
}

// ---------------------------------------------------------------------------
// Kernel 2: Jonker-Volgenant LSA, one wave (32 threads) per batch problem.
// 1-indexed square problem n=128; cols > T have cost 0 (padding). Lane l owns
// columns j with j%32==l (lane 0 additionally owns j=128). All state in LDS.
// ---------------------------------------------------------------------------
__global__ __launch_bounds__(32) void hungarian_kernel(
    const float* __restrict__ Cin,    // [B,Q,T]
    float*       __restrict__ rows_out,
    float*       __restrict__ cols_out)
{
    __shared__ float a[QQ * TT];            // real cost columns
    __shared__ float u[QQ + 2], vv[QQ + 2], minv[QQ + 4];
    __shared__ int   p[QQ + 4], way[QQ + 4], used[QQ + 4];

    const int b    = blockIdx.x;
    const int lane = threadIdx.x;

    const float4* src = (const float4*)(Cin + (size_t)b * QQ * TT);
    float4* dst = (float4*)a;
    for (int i = lane; i < QQ * TT / 4; i += 32) dst[i] = src[i];

    #pragma unroll
    for (int s = 0; s < 5; ++s) {
        const int j = lane + 32 * s;
        if (j <= QQ) { u[j] = 0.0f; vv[j] = 0.0f; p[j] = 0; }
    }
    __syncthreads();

    for (int i = 1; i <= QQ; ++i) {
        if (lane == 0) p[0] = i;
        #pragma unroll
        for (int s = 0; s < 5; ++s) {
            const int j = lane + 32 * s;
            if (j <= QQ) { minv[j] = CINF; used[j] = 0; way[j] = 0; }
        }
        int j0 = 0;
        __syncthreads();

        while (true) {
            const int i0 = p[j0];            // uniform LDS broadcast
            if (i0 == 0) break;
            if (lane == (j0 & 31)) used[j0] = 1;   // owner-lane write
            const float ui0 = u[i0];

            float bestv = CINF;
            int   bestj = 0x7fffffff;
            #pragma unroll
            for (int s = 0; s < 5; ++s) {
                const int j = lane + 32 * s;
                if (j > QQ) continue;
                if (!used[j]) {
                    const float aij = (j >= 1 && j <= TT) ? a[(i0 - 1) * TT + (j - 1)] : 0.0f;
                    const float cur = aij - ui0 - vv[j];
                    if (cur < minv[j]) { minv[j] = cur; way[j] = j0; }
                    const float mv = minv[j];
                    if (j != 0 && (mv < bestv || (mv == bestv && j < bestj))) {
                        bestv = mv; bestj = j;
                    }
                }
            }
            #pragma unroll
            for (int off = 16; off >= 1; off >>= 1) {   // argmin, first-index tiebreak
                const float ov = __shfl_xor(bestv, off, 32);
                const int   oj = __shfl_xor(bestj, off, 32);
                if (ov < bestv || (ov == bestv && oj < bestj)) { bestv = ov; bestj = oj; }
            }
            const float delta = bestv;
            const int   j1    = bestj;

            #pragma unroll
            for (int s = 0; s < 5; ++s) {
                const int j = lane + 32 * s;
                if (j > QQ) continue;
                if (used[j]) { u[p[j]] += delta; vv[j] -= delta; }  // p[j] distinct -> conflict-free
                else          minv[j] -= delta;
            }
            j0 = j1;
            __syncthreads();   // u[] scatter visible before next ui0 read
        }

        __syncthreads();
        if (lane == 0) {       // augmenting-path unwind
            while (j0 != 0) { const int j1 = way[j0]; p[j0] = p[j1]; j0 = j1; }
        }
        __syncthreads();
    }

    // row4col[t] = p[t+1]-1 for real cols; emit sorted by row (argsort rank)
    const int rsel = p[lane + 1] - 1;
    int rank = 0;
    for (int mth = 0; mth < TT; ++mth) {
        const int rm = __shfl(rsel, mth, 32);
        if (rm < rsel || (rm == rsel && mth < lane)) ++rank;
    }
    rows_out[b * TT + rank] = (float)rsel;
    cols_out[b * TT + rank] = (float)lane;
}

// ---------------------------------------------------------------------------
extern "C" void kernel_launch(void* const* d_in, const int* in_sizes, int n_in,
                              void* d_out, int out_size, void* d_ws, size_t ws_size,
                              hipStream_t stream) {
    (void)in_sizes; (void)n_in; (void)d_ws; (void)ws_size; (void)out_size;
    const float* logits = (const float*)d_in[0];
    const float* pboxes = (const float*)d_in[1];
    const int*   labels = (const int*)d_in[2];
    const float* tboxes = (const float*)d_in[3];

    float* out  = (float*)d_out;
    float* Cmat = out;                                   // [B,Q,T]
    float* rows = out + (size_t)BQ * QQ * TT;            // [B,T]
    float* cols = rows + (size_t)BQ * TT;                // [B,T]

    hipLaunchKernelGGL(cost_kernel, dim3(BQ), dim3(256), 0, stream,
                       logits, pboxes, labels, tboxes, Cmat);
    hipLaunchKernelGGL(hungarian_kernel, dim3(BQ), dim3(32), 0, stream,
                       Cmat, rows, cols);
}